// FewSOME_71476845740752
// MI455X (gfx1250) — compile-verified
//
#include <hip/hip_runtime.h>

typedef float v2f __attribute__((ext_vector_type(2)));
typedef float v8f __attribute__((ext_vector_type(8)));

#define LP     8192
#define DIM    128
#define EPSC   1e-6f
#define MARGIN 0.2f
#define DEPS2  (128.0f * 1e-12f)   // d * EPS^2

__device__ __forceinline__ float wave_reduce(float v) {
#pragma unroll
    for (int m = 16; m >= 1; m >>= 1)
        v += __shfl_xor(v, m, 32);
    return v;
}

// ---- Stage 1: L2-normalize P rows; emit normalized rows + per-row sum / sumsq ----
__global__ __launch_bounds__(256) void normalize_p(const float* __restrict__ P,
                                                   float* __restrict__ Pn,
                                                   float* __restrict__ sqP,
                                                   float* __restrict__ sP) {
    int wave = threadIdx.x >> 5, lane = threadIdx.x & 31;
    int row  = blockIdx.x * 8 + wave;
    float4 x = ((const float4*)(P + (size_t)row * DIM))[lane];
    float ss = x.x * x.x + x.y * x.y + x.z * x.z + x.w * x.w;
    ss = wave_reduce(ss);
    float inv = 1.0f / fmaxf(sqrtf(ss), 1e-12f);
    float4 y; y.x = x.x * inv; y.y = x.y * inv; y.z = x.z * inv; y.w = x.w * inv;
    ((float4*)(Pn + (size_t)row * DIM))[lane] = y;
    float s  = y.x + y.y + y.z + y.w;
    float sq = y.x * y.x + y.y * y.y + y.z * y.z + y.w * y.w;
    s  = wave_reduce(s);
    sq = wave_reduce(sq);
    if (lane == 0) { sP[row] = s; sqP[row] = sq; }
}

// ---- Stage 2: normalize n[0]; emit normalized vector + (sumsq, sum) ----
__global__ void normalize_n(const float* __restrict__ Nm,
                            float* __restrict__ nn,
                            float* __restrict__ nstats) {
    int lane = threadIdx.x;
    float4 x = ((const float4*)Nm)[lane];  // row t = 0
    float ss = x.x * x.x + x.y * x.y + x.z * x.z + x.w * x.w;
    ss = wave_reduce(ss);
    float inv = 1.0f / fmaxf(sqrtf(ss), 1e-12f);
    float4 y; y.x = x.x * inv; y.y = x.y * inv; y.z = x.z * inv; y.w = x.w * inv;
    ((float4*)nn)[lane] = y;
    float s  = y.x + y.y + y.z + y.w;
    float sq = y.x * y.x + y.y * y.y + y.z * y.z + y.w * y.w;
    s  = wave_reduce(s);
    sq = wave_reduce(sq);
    if (lane == 0) { nstats[0] = sq; nstats[1] = s; }
}

// ---- Stage 3: d_pn[i] = ||p_i - n0 + EPS||  via expanded form ----
__global__ __launch_bounds__(256) void compute_dpn(const float* __restrict__ Pn,
                                                   const float* __restrict__ sqP,
                                                   const float* __restrict__ sP,
                                                   const float* __restrict__ nn,
                                                   const float* __restrict__ nstats,
                                                   float* __restrict__ dpn) {
    int wave = threadIdx.x >> 5, lane = threadIdx.x & 31;
    int row  = blockIdx.x * 8 + wave;
    float4 a = ((const float4*)(Pn + (size_t)row * DIM))[lane];
    float4 b = ((const float4*)nn)[lane];
    float dot = a.x * b.x + a.y * b.y + a.z * b.z + a.w * b.w;
    dot = wave_reduce(dot);
    if (lane == 0) {
        float sq = sqP[row] + nstats[0] - 2.0f * dot
                 + 2.0f * EPSC * (sP[row] - nstats[1]) + DEPS2;
        dpn[row] = sqrtf(fmaxf(sq, 0.0f));
    }
}

// ---- Stage 4: WMMA fp32 GEMM tile (16x16, K=128) + fused loss epilogue ----
// One wave per 16x16 tile of S = Pn * Pn^T, 32 x V_WMMA_F32_16X16X4_F32.
__global__ __launch_bounds__(256) void fewsome_main(const float* __restrict__ Pn,
                                                    const float* __restrict__ sqP,
                                                    const float* __restrict__ sP,
                                                    const float* __restrict__ dpn,
                                                    float* __restrict__ partial) {
    __shared__ float smem[8];
    int wave   = threadIdx.x >> 5;
    int lane   = threadIdx.x & 31;
    int lane16 = lane & 15;
    int hi     = lane >> 4;
    int rowBase = (blockIdx.y * 8 + wave) * 16;
    int colBase = blockIdx.x * 16;

    // A-layout (16x4 f32): lane holds A[lane16, k0 + 2*hi + {0,1}].
    // B = Pn^T tile (4x16): lane holds Pn[colBase+lane16, k0 + 2*hi + {0,1}].
    const float* arow = Pn + (size_t)(rowBase + lane16) * DIM + 2 * hi;
    const float* brow = Pn + (size_t)(colBase + lane16) * DIM + 2 * hi;

    v8f acc = {};
#pragma unroll
    for (int k = 0; k < DIM; k += 4) {
        v2f a = *(const v2f*)(arow + k);
        v2f b = *(const v2f*)(brow + k);
        acc = __builtin_amdgcn_wmma_f32_16x16x4_f32(
            /*neg_a=*/false, a, /*neg_b=*/false, b,
            /*c_mod=*/(short)0, acc, /*reuse_a=*/false, /*reuse_b=*/false);
    }

    // D layout: acc[r] = S[rowBase + r + 8*hi][colBase + lane16]
    int   j   = colBase + lane16;
    float sqj = sqP[j];
    float sj  = sP[j];
    float lsum = 0.0f;
#pragma unroll
    for (int r = 0; r < 8; ++r) {
        int   i   = rowBase + 8 * hi + r;
        float sqd = sqP[i] + sqj - 2.0f * acc[r]
                  + 2.0f * EPSC * (sP[i] - sj) + DEPS2;
        float dpp = sqrtf(fmaxf(sqd, 0.0f));
        lsum += fmaxf(dpp + MARGIN - dpn[i], 0.0f);
    }
    lsum = wave_reduce(lsum);
    if (lane == 0) smem[wave] = lsum;
    __syncthreads();
    if (threadIdx.x == 0) {
        float t = 0.0f;
#pragma unroll
        for (int w = 0; w < 8; ++w) t += smem[w];
        partial[blockIdx.y * gridDim.x + blockIdx.x] = t;
    }
}

// ---- Stage 5: deterministic final reduction (double accumulation) ----
__global__ __launch_bounds__(256) void final_reduce(const float* __restrict__ partial,
                                                    int n, float* __restrict__ out) {
    __shared__ double smem[256];
    double s = 0.0;
    for (int idx = threadIdx.x; idx < n; idx += 256) s += (double)partial[idx];
    smem[threadIdx.x] = s;
    __syncthreads();
    for (int stride = 128; stride > 0; stride >>= 1) {
        if ((int)threadIdx.x < stride) smem[threadIdx.x] += smem[threadIdx.x + stride];
        __syncthreads();
    }
    if (threadIdx.x == 0) {
        double denom = (double)(LP - 1) * (double)LP;
        float loss = (float)(smem[0] / denom);
        out[0] = fmaxf(loss, 0.0f);
    }
}

extern "C" void kernel_launch(void* const* d_in, const int* in_sizes, int n_in,
                              void* d_out, int out_size, void* d_ws, size_t ws_size,
                              hipStream_t stream) {
    const float* P = (const float*)d_in[0];   // [8192,128]
    const float* N = (const float*)d_in[1];   // [64,128]
    float* out = (float*)d_out;

    float* ws     = (float*)d_ws;
    float* Pn     = ws;               // 8192*128 = 1,048,576
    float* sqP    = Pn + LP * DIM;    // 8192
    float* sP     = sqP + LP;         // 8192
    float* dpn    = sP + LP;          // 8192
    float* nn     = dpn + LP;         // 128
    float* nstats = nn + DIM;         // 2
    float* partial = nstats + 2;      // 512*64 = 32768

    normalize_p<<<LP / 8, 256, 0, stream>>>(P, Pn, sqP, sP);
    normalize_n<<<1, 32, 0, stream>>>(N, nn, nstats);
    compute_dpn<<<LP / 8, 256, 0, stream>>>(Pn, sqP, sP, nn, nstats, dpn);

    dim3 grid(LP / 16, LP / (16 * 8));          // 512 col-tiles x 64 row-groups
    fewsome_main<<<grid, 256, 0, stream>>>(Pn, sqP, sP, dpn, partial);

    final_reduce<<<1, 256, 0, stream>>>(partial, (LP / 16) * (LP / 128), out);
}